// EdgePredictorGNN_18013092839945
// MI455X (gfx1250) — compile-verified
//
#include <hip/hip_runtime.h>
#include <hip/hip_bf16.h>

typedef __attribute__((ext_vector_type(2))) float v2f;
typedef __attribute__((ext_vector_type(8))) float v8f;

#define IN_C 128
#define HID  64
#define OUT_C 16

// ---------------------------------------------------------------------------
// Degree / normalization
// ---------------------------------------------------------------------------
__global__ void deg_init_k(float* __restrict__ deg, int n) {
  int i = blockIdx.x * blockDim.x + threadIdx.x;
  if (i < n) deg[i] = 1.0f;  // self-loop contributes 1 to every node
}

__global__ void deg_edges_k(float* __restrict__ deg, const int* __restrict__ ei, int E) {
  int e = blockIdx.x * blockDim.x + threadIdx.x;
  if (e < E) atomicAdd(&deg[ei[E + e]], 1.0f);  // dst row of edge_index
}

__global__ void dinv_k(float* __restrict__ deg, int n) {
  int i = blockIdx.x * blockDim.x + threadIdx.x;
  if (i < n) {
    float d = deg[i];
    deg[i] = (d > 0.0f) ? rsqrtf(d) : 0.0f;
  }
}

// ---------------------------------------------------------------------------
// Dense GEMM: C[M,N16] = A[M,K] @ B[K,N], one 16-wide N tile per block,
// 4 waves/block each owning a 16-row M tile; B slice staged in LDS (4x reuse).
// fp32 WMMA 16x16x4. Optional fused ReLU on A (consumer-side ReLU fusion).
// A-layout (f32 16x4): lane m=lane&15; half=lane>>4 -> K pair {2h, 2h+1}.
// B-layout (f32 4x16): lane n=lane&15; half -> K pair {2h, 2h+1}.
// C-layout: VGPR r -> row r + 8*half, col lane&15.
// ---------------------------------------------------------------------------
template <bool RELU_A>
__global__ void gemm_wmma_k(const float* __restrict__ A, const float* __restrict__ B,
                            float* __restrict__ C, int M, int K, int N) {
  __shared__ float Bs[IN_C * 16];  // up to K=128 rows x 16 cols, 8 KB

  const int tid  = threadIdx.x;
  const int wave = tid >> 5;
  const int lane = tid & 31;
  const int mr   = lane & 15;
  const int half = lane >> 4;
  const int n0   = blockIdx.y * 16;

  // Stage B[K x 16] slice into LDS (coalesced 64B row chunks).
  for (int i = tid; i < K * 16; i += 128) {
    int kk = i >> 4, nn = i & 15;
    Bs[i] = B[(size_t)kk * N + n0 + nn];
  }
  __syncthreads();

  const int tile = blockIdx.x * 4 + wave;  // wave-uniform guard: EXEC stays all-1s
  if (tile * 16 < M) {
    const int m0 = tile * 16;
    const float* arow = A + (size_t)(m0 + mr) * K + 2 * half;
    const float* brow = Bs + 2 * half * 16 + mr;

    v8f acc = {};
    for (int k = 0; k < K; k += 4) {
      v2f a, b;
      a[0] = arow[k];
      a[1] = arow[k + 1];
      if (RELU_A) { a[0] = fmaxf(a[0], 0.0f); a[1] = fmaxf(a[1], 0.0f); }
      b[0] = brow[k * 16];        // Bs[(k+2h)*16 + mr]
      b[1] = brow[k * 16 + 16];   // Bs[(k+2h+1)*16 + mr]
      acc = __builtin_amdgcn_wmma_f32_16x16x4_f32(false, a, false, b,
                                                  (short)0, acc, false, false);
    }

    float* crow = C + (size_t)(m0 + 8 * half) * N + n0 + mr;
#pragma unroll
    for (int r = 0; r < 8; ++r) crow[(size_t)r * N] = acc[r];
  }
}

// ---------------------------------------------------------------------------
// Aggregation (float4-vectorized):
//   out[i,c] = bias[c] + dinv[i]^2 * h[i,c]              (self-loop init)
//   out[dst,c] += dinv[src]*dinv[dst] * h[src,c]         (edges, f32 atomics)
// ReLU is NOT applied here; it is fused into the consumers.
// ---------------------------------------------------------------------------
__global__ void agg_init_k(float* __restrict__ out, const float* __restrict__ h,
                           const float* __restrict__ dinv, const float* __restrict__ bias,
                           int n4) {  // n4 = N * HID/4
  int i = blockIdx.x * blockDim.x + threadIdx.x;
  if (i < n4) {
    int node = i >> 4;          // HID/4 == 16 quads per node
    int c4   = (i & 15) * 4;
    float d2 = dinv[node]; d2 *= d2;
    const float4 hv = ((const float4*)h)[i];
    const float4 bv = *(const float4*)&bias[c4];
    float4 o;
    o.x = bv.x + d2 * hv.x;
    o.y = bv.y + d2 * hv.y;
    o.z = bv.z + d2 * hv.z;
    o.w = bv.w + d2 * hv.w;
    ((float4*)out)[i] = o;
  }
}

__global__ void agg_edges_k(float* __restrict__ out, const float* __restrict__ h,
                            const float* __restrict__ dinv, const int* __restrict__ ei,
                            int E) {
  int idx = blockIdx.x * blockDim.x + threadIdx.x;  // E * 16 quad-items
  if (idx >= E * 16) return;
  int e  = idx >> 4;
  int c4 = (idx & 15) * 4;
  int s = ei[e];
  int d = ei[E + e];
  float w = dinv[s] * dinv[d];
  const float4 hv = *(const float4*)&h[(size_t)s * HID + c4];
  float* op = &out[(size_t)d * HID + c4];
  atomicAdd(op + 0, w * hv.x);
  atomicAdd(op + 1, w * hv.y);
  atomicAdd(op + 2, w * hv.z);
  atomicAdd(op + 3, w * hv.w);
}

// ---------------------------------------------------------------------------
// Edge MLP: ef = [relu(h[row]) | relu(h[col])] (128),
//           z = relu(ef @ Wm1 + bm1) (64), out = z @ Wm2 + bm2 (16).
// One wave per 16 edges, 4 waves/block. Wm1 staged in LDS with row stride 72
// (2*72 mod 64 == 16 -> the two half-waves hit disjoint bank ranges), Wm2
// staged at natural stride (half-waves differ by 32 banks). z round-trips
// through LDS (stride 68) to convert C-layout -> A-layout.
// ---------------------------------------------------------------------------
#define W1S_STRIDE 72

__global__ void edge_mlp_k(const float* __restrict__ h, const int* __restrict__ ei,
                           const float* __restrict__ Wm1, const float* __restrict__ bm1,
                           const float* __restrict__ Wm2, const float* __restrict__ bm2,
                           float* __restrict__ out, int E) {
  __shared__ float w1s[IN_C * W1S_STRIDE];  // 36.9 KB
  __shared__ float w2s[HID * OUT_C];        //  4 KB
  __shared__ float zbuf[4][16][68];         // 17.4 KB

  const int tid  = threadIdx.x;
  const int wave = tid >> 5;
  const int lane = tid & 31;
  const int mr   = lane & 15;
  const int half = lane >> 4;
  const int e0   = (blockIdx.x * 4 + wave) * 16;

  // Stage Wm1 [128 x 64] -> padded LDS, Wm2 [64 x 16] -> LDS (float4 copies).
  for (int i = tid; i < IN_C * 16; i += 128) {
    int row = i >> 4, c4 = (i & 15) * 4;
    *(float4*)&w1s[row * W1S_STRIDE + c4] = *(const float4*)&Wm1[(size_t)row * HID + c4];
  }
  for (int i = tid; i < HID * OUT_C / 4; i += 128) {
    ((float4*)w2s)[i] = ((const float4*)Wm2)[i];
  }
  __syncthreads();

  const int row = ei[e0 + mr];       // src row of edge_index
  const int col = ei[E + e0 + mr];   // dst row of edge_index
  const float* efr = h + (size_t)row * HID + 2 * half;  // k in [0,64)
  const float* efc = h + (size_t)col * HID + 2 * half;  // k in [64,128)

  v8f acc[4] = {v8f{}, v8f{}, v8f{}, v8f{}};

  // K = 0..63 : A from relu(h[row])
  for (int k = 0; k < 64; k += 4) {
    v2f a;
    a[0] = fmaxf(efr[k], 0.0f);
    a[1] = fmaxf(efr[k + 1], 0.0f);
    const float* bp = w1s + (k + 2 * half) * W1S_STRIDE + mr;
#pragma unroll
    for (int nt = 0; nt < 4; ++nt) {
      v2f b;
      b[0] = bp[nt * 16];
      b[1] = bp[W1S_STRIDE + nt * 16];
      acc[nt] = __builtin_amdgcn_wmma_f32_16x16x4_f32(false, a, false, b,
                                                      (short)0, acc[nt], false, false);
    }
  }
  // K = 64..127 : A from relu(h[col])
  for (int k = 0; k < 64; k += 4) {
    v2f a;
    a[0] = fmaxf(efc[k], 0.0f);
    a[1] = fmaxf(efc[k + 1], 0.0f);
    const float* bp = w1s + (64 + k + 2 * half) * W1S_STRIDE + mr;
#pragma unroll
    for (int nt = 0; nt < 4; ++nt) {
      v2f b;
      b[0] = bp[nt * 16];
      b[1] = bp[W1S_STRIDE + nt * 16];
      acc[nt] = __builtin_amdgcn_wmma_f32_16x16x4_f32(false, a, false, b,
                                                      (short)0, acc[nt], false, false);
    }
  }

  // bias + relu, C-layout -> LDS (row = r + 8*half, col = nt*16 + mr)
#pragma unroll
  for (int nt = 0; nt < 4; ++nt) {
    float bias = bm1[nt * 16 + mr];
#pragma unroll
    for (int r = 0; r < 8; ++r)
      zbuf[wave][r + 8 * half][nt * 16 + mr] = fmaxf(acc[nt][r] + bias, 0.0f);
  }
  __syncthreads();

  // Second GEMM: z(16x64) @ Wm2(64x16), A and B both from LDS
  v8f o = {};
  for (int k = 0; k < 64; k += 4) {
    v2f a, b;
    a[0] = zbuf[wave][mr][k + 2 * half];
    a[1] = zbuf[wave][mr][k + 2 * half + 1];
    b[0] = w2s[(k + 2 * half) * OUT_C + mr];
    b[1] = w2s[(k + 2 * half + 1) * OUT_C + mr];
    o = __builtin_amdgcn_wmma_f32_16x16x4_f32(false, a, false, b,
                                              (short)0, o, false, false);
  }

  float bo = bm2[mr];
  float* op = out + (size_t)(e0 + 8 * half) * OUT_C + mr;
#pragma unroll
  for (int r = 0; r < 8; ++r) op[(size_t)r * OUT_C] = o[r] + bo;
}

// ---------------------------------------------------------------------------
// Launch
// ---------------------------------------------------------------------------
extern "C" void kernel_launch(void* const* d_in, const int* in_sizes, int n_in,
                              void* d_out, int out_size, void* d_ws, size_t ws_size,
                              hipStream_t stream) {
  const float* x   = (const float*)d_in[0];
  const int*   ei  = (const int*)d_in[1];
  const float* W1  = (const float*)d_in[2];
  const float* b1  = (const float*)d_in[3];
  const float* W2  = (const float*)d_in[4];
  const float* b2  = (const float*)d_in[5];
  const float* Wm1 = (const float*)d_in[6];
  const float* bm1 = (const float*)d_in[7];
  const float* Wm2 = (const float*)d_in[8];
  const float* bm2 = (const float*)d_in[9];
  float* out = (float*)d_out;

  const int N = in_sizes[0] / IN_C;  // 50000
  const int E = in_sizes[1] / 2;     // 800000

  // Workspace: dinv [N] | bufA [N*HID] | bufB [N*HID]  (~26 MB)
  float* deg  = (float*)d_ws;
  float* bufA = deg + (size_t)((N + 63) & ~63);
  float* bufB = bufA + (size_t)N * HID;

  const int mtiles = (N + 15) / 16;              // 3125
  const int gx     = (mtiles + 3) / 4;           // 4 waves (m-tiles) per block
  const int n4     = N * (HID / 4);
  const int e4     = E * (HID / 4);
  const int nb  = (N + 255) / 256;
  const int eb  = (E + 255) / 256;
  const int n4b = (n4 + 255) / 256;
  const int e4b = (e4 + 255) / 256;

  // normalization
  deg_init_k<<<nb, 256, 0, stream>>>(deg, N);
  deg_edges_k<<<eb, 256, 0, stream>>>(deg, ei, E);
  dinv_k<<<nb, 256, 0, stream>>>(deg, N);

  // layer 1: h1 = x @ W1 ; a1 = agg(h1) + b1   (ReLU fused into consumers)
  gemm_wmma_k<false><<<dim3(gx, HID / 16), 128, 0, stream>>>(x, W1, bufA, N, IN_C, HID);
  agg_init_k<<<n4b, 256, 0, stream>>>(bufB, bufA, deg, b1, n4);
  agg_edges_k<<<e4b, 256, 0, stream>>>(bufB, bufA, deg, ei, E);

  // layer 2: h2 = relu(a1) @ W2 ; a2 = agg(h2) + b2
  gemm_wmma_k<true><<<dim3(gx, HID / 16), 128, 0, stream>>>(bufB, W2, bufA, N, HID, HID);
  agg_init_k<<<n4b, 256, 0, stream>>>(bufB, bufA, deg, b2, n4);
  agg_edges_k<<<e4b, 256, 0, stream>>>(bufB, bufA, deg, ei, E);

  // edge MLP (applies relu to a2 on load) -> out [E, 16]
  edge_mlp_k<<<E / 64, 128, 0, stream>>>(bufB, ei, Wm1, bm1, Wm2, bm2, out, E);
}